// WeightedFSOhemCELoss_6708738916951
// MI455X (gfx1250) — compile-verified
//
#include <hip/hip_runtime.h>
#include <hip/hip_bf16.h>
#include <math.h>

#define OHEM_THRESH 0.7f
#define THREADS     256
#define HIST1_N     4096   // top 12 bits of float pattern
#define HIST2_N     4096   // middle 12 bits
#define HIST3_N     256    // low 8 bits
#define RBLK        1024   // reduction blocks

// ===========================================================================
// Pass 1 (vectorized): 4 consecutive pixels per thread -> all pred reads are
// global_load_b128, tgt read is b128, prob/loss writes are b128.
// Requires HW % 4 == 0 and NPIX % 4 == 0 (true for 8x19x512x512).
// ===========================================================================
template <int C>
__global__ void k_softmax_hist_v4(const float* __restrict__ pred,
                                  const int*   __restrict__ tgt,
                                  float*       __restrict__ prob,
                                  float*       __restrict__ loss,
                                  unsigned*    __restrict__ hist1,
                                  unsigned*    __restrict__ sc,   // sc[0]=n_valid
                                  int HW, int NPIX)
{
    __shared__ unsigned lh[HIST1_N];
    __shared__ unsigned sv[THREADS];
    for (int i = threadIdx.x; i < HIST1_N; i += blockDim.x) lh[i] = 0;
    __syncthreads();

    unsigned localValid = 0;
    const int NV     = NPIX >> 2;                 // float4 elements
    const int stride = gridDim.x * blockDim.x;
    for (int q = blockIdx.x * blockDim.x + threadIdx.x; q < NV; q += stride) {
        const int p = q << 2;                     // first pixel of the group
        // prefetch next grid-stride tile of the streaming read (global_prefetch_b8)
        int qn = q + stride;
        if (qn < NV) {
            int pn = qn << 2;
            int n2 = pn / HW, hw2 = pn - n2 * HW;
            __builtin_prefetch(&pred[(size_t)n2 * C * HW + hw2], 0, 1);
        }

        int n  = p / HW;                          // group never crosses n (HW%4==0)
        int hw = p - n * HW;
        size_t base = (size_t)n * C * HW + hw;

        const int4 t4 = *(const int4*)(tgt + p);
        int4 tt;
        tt.x = (t4.x == -1) ? 0 : t4.x;
        tt.y = (t4.y == -1) ? 0 : t4.y;
        tt.z = (t4.z == -1) ? 0 : t4.z;
        tt.w = (t4.w == -1) ? 0 : t4.w;

        float4 v[C];
        float4 m  = make_float4(-INFINITY, -INFINITY, -INFINITY, -INFINITY);
        float4 xt = make_float4(0.f, 0.f, 0.f, 0.f);
#pragma unroll
        for (int c = 0; c < C; ++c) {
            float4 x = *(const float4*)(pred + base + (size_t)c * HW);
            v[c] = x;
            m.x = fmaxf(m.x, x.x); m.y = fmaxf(m.y, x.y);
            m.z = fmaxf(m.z, x.z); m.w = fmaxf(m.w, x.w);
            if (c == tt.x) xt.x = x.x;
            if (c == tt.y) xt.y = x.y;
            if (c == tt.z) xt.z = x.z;
            if (c == tt.w) xt.w = x.w;
        }
        float4 s = make_float4(0.f, 0.f, 0.f, 0.f);
#pragma unroll
        for (int c = 0; c < C; ++c) {
            s.x += __expf(v[c].x - m.x); s.y += __expf(v[c].y - m.y);
            s.z += __expf(v[c].z - m.z); s.w += __expf(v[c].w - m.w);
        }
        float4 lp, pr;
        lp.x = xt.x - m.x - __logf(s.x);
        lp.y = xt.y - m.y - __logf(s.y);
        lp.z = xt.z - m.z - __logf(s.z);
        lp.w = xt.w - m.w - __logf(s.w);
        pr.x = (t4.x != -1) ? __expf(lp.x) : 2.0f;
        pr.y = (t4.y != -1) ? __expf(lp.y) : 2.0f;
        pr.z = (t4.z != -1) ? __expf(lp.z) : 2.0f;
        pr.w = (t4.w != -1) ? __expf(lp.w) : 2.0f;

        *(float4*)(prob + p) = pr;
        *(float4*)(loss + p) = make_float4(-lp.x, -lp.y, -lp.z, -lp.w);

        localValid += (t4.x != -1) + (t4.y != -1) + (t4.z != -1) + (t4.w != -1);
        atomicAdd(&lh[__float_as_uint(pr.x) >> 20], 1u);
        atomicAdd(&lh[__float_as_uint(pr.y) >> 20], 1u);
        atomicAdd(&lh[__float_as_uint(pr.z) >> 20], 1u);
        atomicAdd(&lh[__float_as_uint(pr.w) >> 20], 1u);
    }
    __syncthreads();
    for (int i = threadIdx.x; i < HIST1_N; i += blockDim.x) {
        unsigned h = lh[i];
        if (h) atomicAdd(&hist1[i], h);
    }
    sv[threadIdx.x] = localValid;
    __syncthreads();
    for (int off = blockDim.x >> 1; off > 0; off >>= 1) {
        if (threadIdx.x < off) sv[threadIdx.x] += sv[threadIdx.x + off];
        __syncthreads();
    }
    if (threadIdx.x == 0 && sv[0]) atomicAdd(&sc[0], sv[0]);
}

// Generic-C / unaligned fallback (online softmax), scalar path.
__global__ void k_softmax_hist_gen(const float* __restrict__ pred,
                                   const int*   __restrict__ tgt,
                                   float*       __restrict__ prob,
                                   float*       __restrict__ loss,
                                   unsigned*    __restrict__ hist1,
                                   unsigned*    __restrict__ sc,
                                   int C, int HW, int NPIX)
{
    __shared__ unsigned lh[HIST1_N];
    __shared__ unsigned sv[THREADS];
    for (int i = threadIdx.x; i < HIST1_N; i += blockDim.x) lh[i] = 0;
    __syncthreads();
    unsigned localValid = 0;
    const int stride = gridDim.x * blockDim.x;
    for (int p = blockIdx.x * blockDim.x + threadIdx.x; p < NPIX; p += stride) {
        int n = p / HW, hw = p - n * HW;
        size_t base = (size_t)n * C * HW + hw;
        int t = tgt[p];
        bool valid = (t != -1);
        int tt = valid ? t : 0;
        float m = -INFINITY, s = 0.f, xt = 0.f;
        for (int c = 0; c < C; ++c) {
            float x = pred[base + (size_t)c * HW];
            if (x > m) { s = s * __expf(m - x) + 1.f; m = x; }
            else       { s += __expf(x - m); }
            if (c == tt) xt = x;
        }
        float lp = xt - m - __logf(s);
        float pr = valid ? __expf(lp) : 2.0f;
        prob[p] = pr;
        loss[p] = -lp;
        localValid += valid ? 1u : 0u;
        atomicAdd(&lh[__float_as_uint(pr) >> 20], 1u);
    }
    __syncthreads();
    for (int i = threadIdx.x; i < HIST1_N; i += blockDim.x) {
        unsigned h = lh[i];
        if (h) atomicAdd(&hist1[i], h);
    }
    sv[threadIdx.x] = localValid;
    __syncthreads();
    for (int off = blockDim.x >> 1; off > 0; off >>= 1) {
        if (threadIdx.x < off) sv[threadIdx.x] += sv[threadIdx.x + off];
        __syncthreads();
    }
    if (threadIdx.x == 0 && sv[0]) atomicAdd(&sc[0], sv[0]);
}

// ===========================================================================
// Radix-select refinement histograms (vectorized b128 reads of prob).
// pass 0: bits[31:20]==sc[1]            -> bucket bits[19:8]  (4096)
// pass 1: bits[31:8] ==(sc[1]<<12)|sc[3]-> bucket bits[7:0]   (256)
// ===========================================================================
__device__ __forceinline__ void refine_one(unsigned bits, unsigned key, int pass,
                                           unsigned* lh)
{
    if (pass == 0) {
        if ((bits >> 20) == key) atomicAdd(&lh[(bits >> 8) & 0xFFFu], 1u);
    } else {
        if ((bits >> 8) == key)  atomicAdd(&lh[bits & 0xFFu], 1u);
    }
}

__global__ void k_hist_refine(const float* __restrict__ prob, int NPIX,
                              const unsigned* __restrict__ sc,
                              unsigned* __restrict__ histOut, int pass)
{
    __shared__ unsigned lh[HIST1_N];
    const int nb = (pass == 0) ? HIST2_N : HIST3_N;
    for (int i = threadIdx.x; i < nb; i += blockDim.x) lh[i] = 0;
    __syncthreads();
    const unsigned key = (pass == 0) ? sc[1] : ((sc[1] << 12) | sc[3]);
    const int stride = gridDim.x * blockDim.x;
    const int NV = NPIX >> 2;
    for (int q = blockIdx.x * blockDim.x + threadIdx.x; q < NV; q += stride) {
        float4 p4 = *(const float4*)(prob + (q << 2));
        refine_one(__float_as_uint(p4.x), key, pass, lh);
        refine_one(__float_as_uint(p4.y), key, pass, lh);
        refine_one(__float_as_uint(p4.z), key, pass, lh);
        refine_one(__float_as_uint(p4.w), key, pass, lh);
    }
    // scalar tail
    for (int p = (NV << 2) + blockIdx.x * blockDim.x + threadIdx.x; p < NPIX;
         p += stride)
        refine_one(__float_as_uint(prob[p]), key, pass, lh);
    __syncthreads();
    for (int i = threadIdx.x; i < nb; i += blockDim.x) {
        unsigned h = lh[i];
        if (h) atomicAdd(&histOut[i], h);
    }
}

// ===========================================================================
// Scalar scan kernels: locate bucket holding 0-indexed rank k.
// sc: [0]=n_valid [1]=b1 [2]=rank1 [3]=b2 [4]=rank2 [5]=kth_bits [6]=thr(f32)
// ===========================================================================
__global__ void k_select(const unsigned* __restrict__ hist, int nbuckets,
                         unsigned* __restrict__ sc,
                         const int* __restrict__ min_kept, int phase)
{
    if (threadIdx.x != 0 || blockIdx.x != 0) return;
    long long k;
    if (phase == 0) {
        long long nv = (long long)sc[0];
        long long mk = (long long)min_kept[0];
        k = nv - 1;
        if (mk < k) k = mk;
        if (k < 0)  k = 0;
    } else if (phase == 1) {
        k = (long long)sc[2];
    } else {
        k = (long long)sc[4];
    }
    unsigned cum = 0;
    int b = 0;
    for (; b < nbuckets; ++b) {
        unsigned h = hist[b];
        if ((long long)cum + (long long)h > k) break;
        cum += h;
    }
    if (b >= nbuckets) b = nbuckets - 1;
    unsigned rem = (unsigned)(k - (long long)cum);
    if (phase == 0)      { sc[1] = (unsigned)b; sc[2] = rem; }
    else if (phase == 1) { sc[3] = (unsigned)b; sc[4] = rem; }
    else {
        unsigned bits = (sc[1] << 20) | (sc[3] << 8) | (unsigned)b;
        sc[5] = bits;
        float thr = fmaxf(__uint_as_float(bits), OHEM_THRESH);
        ((float*)sc)[6] = thr;
    }
}

// ===========================================================================
// Final selection + deterministic two-stage reduction (b128 reads).
// ===========================================================================
__global__ void k_reduce(const float* __restrict__ prob,
                         const float* __restrict__ loss,
                         const unsigned* __restrict__ sc, int NPIX,
                         float* __restrict__ bsum, unsigned* __restrict__ bcnt)
{
    __shared__ float    sf[THREADS];
    __shared__ unsigned su[THREADS];
    const float thr = ((const float*)sc)[6];
    float s = 0.f; unsigned c = 0;
    const int stride = gridDim.x * blockDim.x;
    const int NV = NPIX >> 2;
    for (int q = blockIdx.x * blockDim.x + threadIdx.x; q < NV; q += stride) {
        float4 pr = *(const float4*)(prob + (q << 2));
        float4 ls = *(const float4*)(loss + (q << 2));
        if (pr.x < thr) { s += ls.x; ++c; }
        if (pr.y < thr) { s += ls.y; ++c; }
        if (pr.z < thr) { s += ls.z; ++c; }
        if (pr.w < thr) { s += ls.w; ++c; }
    }
    for (int p = (NV << 2) + blockIdx.x * blockDim.x + threadIdx.x; p < NPIX;
         p += stride) {
        float pr = prob[p];
        if (pr < thr) { s += loss[p]; ++c; }
    }
    sf[threadIdx.x] = s; su[threadIdx.x] = c;
    __syncthreads();
    for (int off = blockDim.x >> 1; off > 0; off >>= 1) {
        if (threadIdx.x < off) {
            sf[threadIdx.x] += sf[threadIdx.x + off];
            su[threadIdx.x] += su[threadIdx.x + off];
        }
        __syncthreads();
    }
    if (threadIdx.x == 0) { bsum[blockIdx.x] = sf[0]; bcnt[blockIdx.x] = su[0]; }
}

__global__ void k_final(const float* __restrict__ bsum,
                        const unsigned* __restrict__ bcnt,
                        int nblk, float* __restrict__ out)
{
    __shared__ float    sf[THREADS];
    __shared__ unsigned su[THREADS];
    float s = 0.f; unsigned c = 0;
    for (int i = threadIdx.x; i < nblk; i += blockDim.x) { s += bsum[i]; c += bcnt[i]; }
    sf[threadIdx.x] = s; su[threadIdx.x] = c;
    __syncthreads();
    for (int off = blockDim.x >> 1; off > 0; off >>= 1) {
        if (threadIdx.x < off) {
            sf[threadIdx.x] += sf[threadIdx.x + off];
            su[threadIdx.x] += su[threadIdx.x + off];
        }
        __syncthreads();
    }
    if (threadIdx.x == 0) out[0] = sf[0] / fmaxf((float)su[0], 1.0f);
}

// ===========================================================================
// Host launcher. ws layout:
//   [0, 4N)   prob f32[N] | [4N, 8N) loss f32[N] |
//   [8N, ..)  hist1 u32[4096] | hist2 u32[4096] | hist3 u32[256] |
//             sc u32[8] | bsum f32[RBLK] | bcnt u32[RBLK]
// ===========================================================================
extern "C" void kernel_launch(void* const* d_in, const int* in_sizes, int n_in,
                              void* d_out, int out_size, void* d_ws, size_t ws_size,
                              hipStream_t stream)
{
    const float* pred     = (const float*)d_in[0];
    const int*   tgt      = (const int*)d_in[1];
    const int*   min_kept = (const int*)d_in[2];

    const int total = in_sizes[0];
    const int NPIX  = in_sizes[1];
    const int C     = total / NPIX;            // 19
    int HW = 512 * 512;                        // (n, c, h, w) layout from reference
    if (HW > NPIX || (NPIX % HW) != 0) HW = NPIX;

    char* ws = (char*)d_ws;
    float*    prob  = (float*)ws;
    float*    loss  = (float*)(ws + (size_t)NPIX * 4);
    unsigned* hist1 = (unsigned*)(ws + (size_t)NPIX * 8);
    unsigned* hist2 = hist1 + HIST1_N;
    unsigned* hist3 = hist2 + HIST2_N;
    unsigned* sc    = hist3 + HIST3_N;
    float*    bsum  = (float*)(sc + 8);
    unsigned* bcnt  = (unsigned*)(bsum + RBLK);

    hipMemsetAsync(hist1, 0, (size_t)(HIST1_N + HIST2_N + HIST3_N + 8) * 4, stream);

    const bool vec_ok = (C == 19) && ((HW & 3) == 0) && ((NPIX & 3) == 0);
    if (vec_ok) {
        int nv = NPIX >> 2;
        int blocks1 = (nv + THREADS - 1) / THREADS;
        if (blocks1 > 2048) blocks1 = 2048;    // 1 float4 group / thread at 2M px
        k_softmax_hist_v4<19><<<blocks1, THREADS, 0, stream>>>(
            pred, tgt, prob, loss, hist1, sc, HW, NPIX);
    } else {
        int blocks1 = (NPIX + THREADS - 1) / THREADS;
        if (blocks1 > 2048) blocks1 = 2048;
        k_softmax_hist_gen<<<blocks1, THREADS, 0, stream>>>(
            pred, tgt, prob, loss, hist1, sc, C, HW, NPIX);
    }

    // exact radix select of the k-th smallest prob (k = min(min_kept, n_valid-1))
    k_select<<<1, 1, 0, stream>>>(hist1, HIST1_N, sc, min_kept, 0);

    int blocks2 = ((NPIX >> 2) + THREADS - 1) / THREADS;
    if (blocks2 > 1024) blocks2 = 1024;
    if (blocks2 < 1)    blocks2 = 1;
    k_hist_refine<<<blocks2, THREADS, 0, stream>>>(prob, NPIX, sc, hist2, 0);
    k_select<<<1, 1, 0, stream>>>(hist2, HIST2_N, sc, min_kept, 1);

    k_hist_refine<<<blocks2, THREADS, 0, stream>>>(prob, NPIX, sc, hist3, 1);
    k_select<<<1, 1, 0, stream>>>(hist3, HIST3_N, sc, min_kept, 2);

    // deterministic selected-loss mean
    k_reduce<<<RBLK, THREADS, 0, stream>>>(prob, loss, sc, NPIX, bsum, bcnt);
    k_final<<<1, THREADS, 0, stream>>>(bsum, bcnt, RBLK, (float*)d_out);
}